// InvariantPointAttention_3DROPE_replacea_84464826843875
// MI455X (gfx1250) — compile-verified
//
#include <hip/hip_runtime.h>
#include <math.h>

typedef __attribute__((ext_vector_type(2))) float v2f;
typedef __attribute__((ext_vector_type(8))) float v8f;

// Problem constants
#define NN   512
#define CSD  384
#define CZD  128
#define HH   12
#define GG   16
#define CHD  48      // 3*G
#define PQKD 4
#define PVD  8

__device__ __forceinline__ v8f wmma4(v2f a, v2f b, v8f c) {
  // V_WMMA_F32_16X16X4_F32 : D = A(16x4) * B(4x16) + C(16x16), f32 throughout
  return __builtin_amdgcn_wmma_f32_16x16x4_f32(false, a, false, b, (short)0, c,
                                               false, false);
}

// ---------------------------------------------------------------------------
// Generic f32 WMMA GEMM:  C[M x Nc] = A[M x K] * op(W) + bias
//   wKN == 0 : W is (Nc x K) row-major  (compute A @ W^T — torch Linear)
//   wKN == 1 : W is (K x Nc) row-major  (compute A @ W)
// One wave per 16x16 tile; 4 waves (128 thr) per block, M mult of 64,
// K mult of 4, Nc mult of 16. blockIdx.z = batch with given strides.
// ---------------------------------------------------------------------------
__global__ __launch_bounds__(128)
void gemm_f32_wmma(const float* __restrict__ A, const float* __restrict__ W,
                   const float* __restrict__ bias, float* __restrict__ C,
                   int M, int K, int Nc, int wKN,
                   long long sA, long long sW, long long sC)
{
  const int bz = blockIdx.z;
  A += (long long)bz * sA;
  W += (long long)bz * sW;
  C += (long long)bz * sC;
  const int wave = threadIdx.x >> 5;
  const int lane = threadIdx.x & 31;
  const int tm = blockIdx.x * 4 + wave;      // 16-row tile index
  const int tn = blockIdx.y;                 // 16-col tile index
  const int arow = tm * 16 + (lane & 15);
  const int bcol = tn * 16 + (lane & 15);
  const int kh   = (lane >> 4) << 1;         // K sub-offset: 0 (lo half), 2 (hi half)
  v8f acc = {};
  const float* ap = A + (size_t)arow * K + kh;
  if (wKN == 0) {
    const float* wp = W + (size_t)bcol * K + kh;
    for (int k0 = 0; k0 < K; k0 += 4) {
      v2f a2 = *(const v2f*)(ap + k0);
      v2f b2 = *(const v2f*)(wp + k0);
      acc = wmma4(a2, b2, acc);
    }
  } else {
    const float* wp = W + (size_t)kh * Nc + bcol;
    for (int k0 = 0; k0 < K; k0 += 4) {
      v2f a2 = *(const v2f*)(ap + k0);
      v2f b2;
      b2[0] = wp[(size_t)k0 * Nc];
      b2[1] = wp[(size_t)(k0 + 1) * Nc];
      acc = wmma4(a2, b2, acc);
    }
  }
  const float bv = bias ? bias[bcol] : 0.0f;
  const int r0 = (lane >> 4) * 8;
  #pragma unroll
  for (int r = 0; r < 8; ++r) {
    int rr = tm * 16 + r0 + r;
    C[(size_t)rr * Nc + bcol] = acc[r] + bv;
  }
}

// ---------------------------------------------------------------------------
// Specialized fused z projection:  zout[262144 x 48] = z[262144 x 128] @ Wzc^T
// Wzc = [wb(12) ; wdz(32) ; zeros(4)]  staged in LDS once per block.
// One wave -> 16 rows x all 48 cols (3 accumulators): per K-step
// 1 global A load feeds 3 WMMAs (B from LDS). This is the roofline kernel
// (134 MB stream) — maximize loads-in-flight per issued instruction.
// ---------------------------------------------------------------------------
__global__ __launch_bounds__(128)
void zproj_kernel(const float* __restrict__ Z, const float* __restrict__ Wzc,
                  const float* __restrict__ bzc, float* __restrict__ zout)
{
  __shared__ float Wl[48 * 128];
  __shared__ float Bl[48];
  const int tid = threadIdx.x;
  for (int i = tid; i < 48 * 128; i += 128) Wl[i] = Wzc[i];
  if (tid < 48) Bl[tid] = bzc[tid];
  __syncthreads();
  const int wave = tid >> 5, lane = tid & 31;
  const int tm   = blockIdx.x * 4 + wave;
  const int arow = tm * 16 + (lane & 15);
  const int kh   = (lane >> 4) << 1;
  const int cl   = lane & 15;
  const float* ap = Z + (size_t)arow * 128 + kh;
  v8f acc0 = {}, acc1 = {}, acc2 = {};
  #pragma unroll 4
  for (int k0 = 0; k0 < 128; k0 += 4) {
    v2f a2 = *(const v2f*)(ap + k0);
    v2f b0 = *(const v2f*)(&Wl[(size_t)(cl)      * 128 + kh + k0]);
    v2f b1 = *(const v2f*)(&Wl[(size_t)(cl + 16) * 128 + kh + k0]);
    v2f b2 = *(const v2f*)(&Wl[(size_t)(cl + 32) * 128 + kh + k0]);
    acc0 = wmma4(a2, b0, acc0);
    acc1 = wmma4(a2, b1, acc1);
    acc2 = wmma4(a2, b2, acc2);
  }
  const int r0 = (lane >> 4) * 8;
  const float bv0 = Bl[cl], bv1 = Bl[cl + 16], bv2 = Bl[cl + 32];
  #pragma unroll
  for (int r = 0; r < 8; ++r) {
    size_t rr = (size_t)(tm * 16 + r0 + r) * 48;
    zout[rr + cl]      = acc0[r] + bv0;
    zout[rr + 16 + cl] = acc1[r] + bv1;
    zout[rr + 32 + cl] = acc2[r] + bv2;
  }
}

// ---------------------------------------------------------------------------
// Prep: fused z-projection weight [48 x 128] = [wb(12); wdz(32); zeros(4)],
// softplus'd head/group weights with their constant factors folded in.
// ---------------------------------------------------------------------------
__global__ __launch_bounds__(64)
void prep_kernel(const float* __restrict__ wb,  const float* __restrict__ bb,
                 const float* __restrict__ wdz, const float* __restrict__ bdz,
                 const float* __restrict__ hwin, const float* __restrict__ gwin,
                 float* __restrict__ Wzc, float* __restrict__ bzc,
                 float* __restrict__ gw, float* __restrict__ hw)
{
  const int tid = threadIdx.x;
  for (int r = tid; r < 48; r += 64) {
    for (int k = 0; k < 128; ++k) {
      float v = (r < 12) ? wb[r * 128 + k]
              : (r < 44) ? wdz[(r - 12) * 128 + k] : 0.0f;
      Wzc[r * 128 + k] = v;
    }
    bzc[r] = (r < 12) ? bb[r] : (r < 44) ? bdz[r - 12] : 0.0f;
  }
  if (tid < 16) gw[tid] = log1pf(expf(gwin[tid])) * 0.25f;                 // * sqrt(1/G)
  if (tid < 12) hw[tid] = log1pf(expf(hwin[tid])) * 0.13608276348795434f;  // * sqrt(1/54)
}

// ---------------------------------------------------------------------------
// Per-residue feature build (block = one n):
//  QF[h][n][0:48]  = gw[g]/sqrt(3) * (R_n @ q_group)      (scaled q_rot)
//  QF[h][n][48:60] = hw[h] * q_pts                        KF analogous, unscaled
//  qb/kb           = -0.5*hw*sum|pts|^2   (row/col bias of point attention)
//  Vn[h][m][0:48]  = v ; Vn[h][m][48:72] = v_pts ; cols 72:80 zero  (K x Nc)
// ---------------------------------------------------------------------------
__global__ __launch_bounds__(128)
void features_kernel(const float* __restrict__ Pq,  const float* __restrict__ Pkv,
                     const float* __restrict__ Pqp, const float* __restrict__ Pkvp,
                     const float* __restrict__ rots, const float* __restrict__ trans,
                     const float* __restrict__ gw,  const float* __restrict__ hw,
                     float* __restrict__ QF, float* __restrict__ KF,
                     float* __restrict__ qb, float* __restrict__ kb,
                     float* __restrict__ Vn)
{
  const int n = blockIdx.x, tid = threadIdx.x;
  __shared__ float R[9], T[3];
  if (tid < 9) R[tid] = rots[n * 9 + tid];
  if (tid < 3) T[tid] = trans[n * 3 + tid];
  __syncthreads();
  const float RS3 = 0.57735026918962576f;  // 1/sqrt(3)

  // rotated q/k groups (scalar part of logits)
  for (int it = tid; it < HH * GG; it += 128) {
    int h = it >> 4, g = it & 15;
    const float* qsrc = Pq  + (size_t)n * 576  + h * 48 + g * 3;
    const float* ksrc = Pkv + (size_t)n * 1152 + h * 96 + g * 3;
    float q0 = qsrc[0], q1 = qsrc[1], q2 = qsrc[2];
    float k0 = ksrc[0], k1 = ksrc[1], k2 = ksrc[2];
    float sc = gw[g] * RS3;
    size_t off = ((size_t)(h * 512 + n)) * 64 + g * 3;
    #pragma unroll
    for (int i = 0; i < 3; ++i) {
      float qr = R[i*3+0]*q0 + R[i*3+1]*q1 + R[i*3+2]*q2;
      float kr = R[i*3+0]*k0 + R[i*3+1]*k1 + R[i*3+2]*k2;
      QF[off + i] = qr * sc;
      KF[off + i] = kr;
    }
  }
  // query/key points (global frame)
  for (int it = tid; it < HH * PQKD; it += 128) {
    int h = it >> 2, p = it & 3;
    float a0 = Pqp[(size_t)n*144 + 0*48  + h*4 + p];
    float a1 = Pqp[(size_t)n*144 + 1*48  + h*4 + p];
    float a2 = Pqp[(size_t)n*144 + 2*48  + h*4 + p];
    float b0 = Pkvp[(size_t)n*432 + 0*144 + h*12 + p];
    float b1 = Pkvp[(size_t)n*432 + 1*144 + h*12 + p];
    float b2 = Pkvp[(size_t)n*432 + 2*144 + h*12 + p];
    float hwv = hw[h];
    size_t off = ((size_t)(h * 512 + n)) * 64 + 48 + p * 3;
    #pragma unroll
    for (int i = 0; i < 3; ++i) {
      float qp = R[i*3+0]*a0 + R[i*3+1]*a1 + R[i*3+2]*a2 + T[i];
      float kp = R[i*3+0]*b0 + R[i*3+1]*b1 + R[i*3+2]*b2 + T[i];
      QF[off + i] = hwv * qp;
      KF[off + i] = kp;
    }
  }
  // zero pad cols 60..63
  for (int it = tid; it < HH * 4; it += 128) {
    int h = it >> 2, j = it & 3;
    size_t off = ((size_t)(h * 512 + n)) * 64 + 60 + j;
    QF[off] = 0.0f; KF[off] = 0.0f;
  }
  // point-norm biases
  for (int h = tid; h < HH; h += 128) {
    float qs = 0.f, ks = 0.f;
    for (int p = 0; p < PQKD; ++p) {
      float a0 = Pqp[(size_t)n*144 + 0*48  + h*4 + p];
      float a1 = Pqp[(size_t)n*144 + 1*48  + h*4 + p];
      float a2 = Pqp[(size_t)n*144 + 2*48  + h*4 + p];
      float b0 = Pkvp[(size_t)n*432 + 0*144 + h*12 + p];
      float b1 = Pkvp[(size_t)n*432 + 1*144 + h*12 + p];
      float b2 = Pkvp[(size_t)n*432 + 2*144 + h*12 + p];
      #pragma unroll
      for (int i = 0; i < 3; ++i) {
        float qp = R[i*3+0]*a0 + R[i*3+1]*a1 + R[i*3+2]*a2 + T[i];
        float kp = R[i*3+0]*b0 + R[i*3+1]*b1 + R[i*3+2]*b2 + T[i];
        qs += qp * qp; ks += kp * kp;
      }
    }
    qb[h * 512 + n] = -0.5f * hw[h] * qs;
    kb[h * 512 + n] = -0.5f * hw[h] * ks;
  }
  // V concat (K x Nc layout for the AV GEMM): v (48) | v_pts (24) | zero (8)
  for (int it = tid; it < HH * 72; it += 128) {
    int h = it / 72, c = it % 72;
    float val;
    if (c < 48) {
      val = Pkv[(size_t)n * 1152 + h * 96 + 48 + c];
    } else {
      int c2 = c - 48, p = c2 / 3, i = c2 % 3;
      float b0 = Pkvp[(size_t)n*432 + 0*144 + h*12 + 4 + p];
      float b1 = Pkvp[(size_t)n*432 + 1*144 + h*12 + 4 + p];
      float b2 = Pkvp[(size_t)n*432 + 2*144 + h*12 + 4 + p];
      val = R[i*3+0]*b0 + R[i*3+1]*b1 + R[i*3+2]*b2 + T[i];
    }
    Vn[((size_t)(h * 512 + n)) * 80 + c] = val;
  }
  for (int it = tid; it < HH * 8; it += 128) {
    int h = it >> 3, r = it & 7;
    Vn[((size_t)(h * 512 + n)) * 80 + 72 + r] = 0.0f;
  }
}

// ---------------------------------------------------------------------------
// Attention: per (16-row strip, head): WMMA QK^T (K=64, A-fragment held in
// registers across all column tiles) -> + biases/bmat/mask -> row softmax in
// LDS -> probs[h][n][m]
// ---------------------------------------------------------------------------
__global__ __launch_bounds__(256)
void attention_kernel(const float* __restrict__ QF, const float* __restrict__ KF,
                      const float* __restrict__ qb, const float* __restrict__ kb,
                      const float* __restrict__ zout, const float* __restrict__ mask,
                      float* __restrict__ probs)
{
  __shared__ float S[16][512];
  __shared__ float red[16][17];
  const int h  = blockIdx.y;
  const int n0 = blockIdx.x << 4;
  const int wave = threadIdx.x >> 5, lane = threadIdx.x & 31;
  const int kh = (lane >> 4) << 1;
  const float* ap = QF + ((size_t)((h << 9) + n0 + (lane & 15))) * 64 + kh;
  v2f afrag[16];
  #pragma unroll
  for (int i = 0; i < 16; ++i) afrag[i] = *(const v2f*)(ap + i * 4);
  for (int ct = wave; ct < 32; ct += 8) {
    const int m0 = ct << 4;
    const float* bp = KF + ((size_t)((h << 9) + m0 + (lane & 15))) * 64 + kh;
    v8f acc = {};
    #pragma unroll
    for (int i = 0; i < 16; ++i) {
      v2f b2 = *(const v2f*)(bp + i * 4);
      acc = wmma4(afrag[i], b2, acc);
    }
    const int r0 = (lane >> 4) * 8;
    const int cl = lane & 15;
    #pragma unroll
    for (int r = 0; r < 8; ++r) S[r0 + r][m0 + cl] = acc[r];
  }
  __syncthreads();

  const int row = threadIdx.x >> 4;
  const int t16 = threadIdx.x & 15;
  const int gn  = n0 + row;
  const float rq   = qb[(h << 9) + gn];
  const float mrow = mask[gn];
  const float SQ13 = 0.57735026918962576f;
  float mx = -3.4e38f;
  for (int c = t16; c < 512; c += 16) {
    float v = S[row][c] + rq + kb[(h << 9) + c]
            + SQ13 * zout[(((size_t)gn << 9) + c) * 48 + h]
            + 100000.0f * (mrow * mask[c] - 1.0f);
    S[row][c] = v;
    mx = fmaxf(mx, v);
  }
  red[row][t16] = mx;
  __syncthreads();
  if (t16 == 0) {
    float m = red[row][0];
    for (int i = 1; i < 16; ++i) m = fmaxf(m, red[row][i]);
    red[row][16] = m;
  }
  __syncthreads();
  const float rmax = red[row][16];
  float sum = 0.f;
  for (int c = t16; c < 512; c += 16) {
    float e = expf(S[row][c] - rmax);
    S[row][c] = e;
    sum += e;
  }
  red[row][t16] = sum;
  __syncthreads();
  if (t16 == 0) {
    float s2 = 0.f;
    for (int i = 0; i < 16; ++i) s2 += red[row][i];
    red[row][16] = s2;
  }
  __syncthreads();
  const float inv = 1.0f / red[row][16];
  float* prow = probs + ((size_t)((h << 9) + gn)) * 512;
  for (int c = t16; c < 512; c += 16) prow[c] = S[row][c] * inv;
}

// ---------------------------------------------------------------------------
// o_pair[n][h][d] = sum_m probs[h][n][m] * pair_z[n][m][d]  (pair cols of zout)
// block = one n (12 waves: h = wave, d = lane); probs rows staged in LDS.
// ---------------------------------------------------------------------------
__global__ __launch_bounds__(384)
void opair_kernel(const float* __restrict__ probs, const float* __restrict__ zout,
                  float* __restrict__ OPAIR)
{
  const int n = blockIdx.x, tid = threadIdx.x;
  __shared__ float Arow[HH * 512];
  for (int i = tid; i < HH * 512; i += 384) {
    int h = i >> 9, m = i & 511;
    Arow[i] = probs[(((size_t)(h << 9) + n) << 9) + m];
  }
  __syncthreads();
  const int h = tid >> 5, d = tid & 31;
  const float* zp = zout + ((size_t)n << 9) * 48 + 12 + d;
  const float* ar = Arow + (h << 9);
  float acc = 0.f;
  #pragma unroll 4
  for (int m = 0; m < 512; ++m) acc += ar[m] * zp[(size_t)m * 48];
  OPAIR[(size_t)n * 384 + (h << 5) + d] = acc;
}

// ---------------------------------------------------------------------------
// Finalize: inverse-rotate o_pt, norms, assemble cat[1344] per residue.
// cat = [o(576) | x(96) | y(96) | z(96) | norms(96) | o_pair(384)]
// ---------------------------------------------------------------------------
__global__ __launch_bounds__(128)
void finalize_kernel(const float* __restrict__ O, const float* __restrict__ OPAIR,
                     const float* __restrict__ rots, const float* __restrict__ trans,
                     float* __restrict__ CAT)
{
  const int n = blockIdx.x, tid = threadIdx.x;
  __shared__ float R[9], T[3];
  if (tid < 9) R[tid] = rots[n * 9 + tid];
  if (tid < 3) T[tid] = trans[n * 3 + tid];
  __syncthreads();
  float* cat = CAT + (size_t)n * 1344;
  for (int i = tid; i < 576; i += 128) {
    int h = i / 48, c = i % 48;
    cat[i] = O[((size_t)(h * 512 + n)) * 80 + c];
  }
  for (int i = tid; i < 96; i += 128) {
    int h = i >> 3, p = i & 7;
    const float* op = O + ((size_t)(h * 512 + n)) * 80 + 48 + p * 3;
    float p0 = op[0] - T[0], p1 = op[1] - T[1], p2 = op[2] - T[2];
    float f0 = R[0]*p0 + R[3]*p1 + R[6]*p2;   // out[i] = sum_j R[j][i]*p[j]
    float f1 = R[1]*p0 + R[4]*p1 + R[7]*p2;
    float f2 = R[2]*p0 + R[5]*p1 + R[8]*p2;
    cat[576 + i] = f0;
    cat[672 + i] = f1;
    cat[768 + i] = f2;
    cat[864 + i] = sqrtf(f0*f0 + f1*f1 + f2*f2 + 1e-8f);
  }
  for (int i = tid; i < 384; i += 128) cat[960 + i] = OPAIR[(size_t)n * 384 + i];
}

// ---------------------------------------------------------------------------
// Workspace layout (float offsets)
// ---------------------------------------------------------------------------
#define OFF_PQ    0ull
#define OFF_PKV   294912ull
#define OFF_PQP   884736ull
#define OFF_PKVP  958464ull
#define OFF_WZC   1179648ull
#define OFF_BZC   1185792ull
#define OFF_GW    1185856ull
#define OFF_HW    1185872ull
#define OFF_QF    1185920ull
#define OFF_KF    1579136ull
#define OFF_QB    1972352ull
#define OFF_KB    1978496ull
#define OFF_VN    1984640ull
#define OFF_ZOUT  2476160ull
#define OFF_PROBS 15059072ull
#define OFF_O     18204800ull
#define OFF_OPAIR 18696320ull
#define OFF_CAT   18892928ull
// total 19581056 floats = ~78.3 MB

extern "C" void kernel_launch(void* const* d_in, const int* in_sizes, int n_in,
                              void* d_out, int out_size, void* d_ws, size_t ws_size,
                              hipStream_t stream) {
  (void)in_sizes; (void)n_in; (void)out_size; (void)ws_size;
  const float* s     = (const float*)d_in[0];
  const float* z     = (const float*)d_in[1];
  const float* rots  = (const float*)d_in[2];
  const float* trans = (const float*)d_in[3];
  const float* mask  = (const float*)d_in[4];
  const float* wq    = (const float*)d_in[5];
  const float* bq    = (const float*)d_in[6];
  const float* wkv   = (const float*)d_in[7];
  const float* bkv   = (const float*)d_in[8];
  const float* wqp   = (const float*)d_in[9];
  const float* bqp   = (const float*)d_in[10];
  const float* wkvp  = (const float*)d_in[11];
  const float* bkvp  = (const float*)d_in[12];
  const float* wb    = (const float*)d_in[13];
  const float* bb    = (const float*)d_in[14];
  const float* wdz   = (const float*)d_in[15];
  const float* bdz   = (const float*)d_in[16];
  const float* hwts  = (const float*)d_in[17];
  const float* gwts  = (const float*)d_in[18];
  const float* wout  = (const float*)d_in[19];
  const float* bout  = (const float*)d_in[20];

  float* ws    = (float*)d_ws;
  float* Pq    = ws + OFF_PQ;
  float* Pkv   = ws + OFF_PKV;
  float* Pqp   = ws + OFF_PQP;
  float* Pkvp  = ws + OFF_PKVP;
  float* Wzc   = ws + OFF_WZC;
  float* bzc   = ws + OFF_BZC;
  float* gw    = ws + OFF_GW;
  float* hw    = ws + OFF_HW;
  float* QF    = ws + OFF_QF;
  float* KF    = ws + OFF_KF;
  float* qb    = ws + OFF_QB;
  float* kb    = ws + OFF_KB;
  float* Vn    = ws + OFF_VN;
  float* zout  = ws + OFF_ZOUT;
  float* probs = ws + OFF_PROBS;
  float* O     = ws + OFF_O;
  float* OPAIR = ws + OFF_OPAIR;
  float* CAT   = ws + OFF_CAT;
  float* out   = (float*)d_out;

  const dim3 blk128(128);

  // 1) prep fused z weights + softplus'd scales
  prep_kernel<<<1, 64, 0, stream>>>(wb, bb, wdz, bdz, hwts, gwts, Wzc, bzc, gw, hw);

  // 2) projections from s: x @ W^T + b  (wKN = 0)
  gemm_f32_wmma<<<dim3(8, 36, 1), blk128, 0, stream>>>(s, wq,   bq,   Pq,   512, 384, 576,  0, 0, 0, 0);
  gemm_f32_wmma<<<dim3(8, 72, 1), blk128, 0, stream>>>(s, wkv,  bkv,  Pkv,  512, 384, 1152, 0, 0, 0, 0);
  gemm_f32_wmma<<<dim3(8, 9,  1), blk128, 0, stream>>>(s, wqp,  bqp,  Pqp,  512, 384, 144,  0, 0, 0, 0);
  gemm_f32_wmma<<<dim3(8, 27, 1), blk128, 0, stream>>>(s, wkvp, bkvp, Pkvp, 512, 384, 432,  0, 0, 0, 0);

  // 3) fused z projection: one pass over the 134 MB z tensor -> [bmat(12)|pair_z(32)|pad(4)]
  //    specialized kernel: LDS-resident weights, 3 accumulators per wave.
  zproj_kernel<<<4096, blk128, 0, stream>>>(z, Wzc, bzc, zout);

  // 4) per-residue features (rotations, scale folding, V concat)
  features_kernel<<<512, blk128, 0, stream>>>(Pq, Pkv, Pqp, Pkvp, rots, trans,
                                              gw, hw, QF, KF, qb, kb, Vn);

  // 5) logits + softmax per (row-strip, head)
  attention_kernel<<<dim3(32, 12, 1), 256, 0, stream>>>(QF, KF, qb, kb, zout, mask, probs);

  // 6) batched probs @ [V | V_pts] per head (wKN = 1)
  gemm_f32_wmma<<<dim3(8, 5, 12), blk128, 0, stream>>>(probs, Vn, nullptr, O,
                                                       512, 512, 80, 1,
                                                       (long long)512 * 512,
                                                       (long long)512 * 80,
                                                       (long long)512 * 80);

  // 7) o_pair from pair cols of fused z output (L2-resident)
  opair_kernel<<<512, 384, 0, stream>>>(probs, zout, OPAIR);

  // 8) inverse rotation, norms, concat
  finalize_kernel<<<512, blk128, 0, stream>>>(O, OPAIR, rots, trans, CAT);

  // 9) output projection: cat @ wout^T + bout
  gemm_f32_wmma<<<dim3(8, 24, 1), blk128, 0, stream>>>(CAT, wout, bout, out,
                                                       512, 1344, 384, 0, 0, 0, 0);
}